// DR2FWL2Kernel_3058016715249
// MI455X (gfx1250) — compile-verified
//
#include <hip/hip_runtime.h>

typedef float v2f __attribute__((ext_vector_type(2)));
typedef float v8f __attribute__((ext_vector_type(8)));

#define XP 68    // LDS pitch (floats) for 64-wide row tiles: 272B rows, 16B aligned, conflict-free b64 reads
#define WPI 160  // LDS pitch (floats) per interleaved K-pair row of W: ≡32 mod 64 banks -> conflict-free b64

static __device__ __forceinline__ v8f wmma4(v2f a, v2f b, v8f c) {
  // V_WMMA_F32_16X16X4_F32 : D(16x16 f32) = A(16x4 f32) * B(4x16 f32) + C
  return __builtin_amdgcn_wmma_f32_16x16x4_f32(false, a, false, b, (short)0, c, false, false);
}

// W global row-major [64][64] -> LDS K-pair interleaved: sW[(k>>1)*WPI + c*2 + (k&1)]
// B fragment for K-base kk (even): {W[kk][col], W[kk+1][col]} = one aligned 8B LDS word.
static __device__ __forceinline__ void load_w_interleaved(
    float* __restrict__ sWp, const float* __restrict__ W, int tid)
{
  for (int k = tid; k < 4096; k += 256) {
    int r = k >> 6, c = k & 63;
    sWp[(r >> 1) * WPI + c * 2 + (r & 1)] = W[k];
  }
}

// -------- fused 2-layer MLP: Y = relu(X@W1 + b1) @ W2 + b2, X:(n,64) row-major --------
__global__ __launch_bounds__(256) void mlp2_kernel(
    const float* __restrict__ X, int n,
    const float* __restrict__ W1, const float* __restrict__ B1,
    const float* __restrict__ W2, const float* __restrict__ B2,
    float* __restrict__ Y)
{
  __shared__ float sX[128 * XP];   // 34816 B: X tile, then per-wave hidden
  __shared__ float sW[32 * WPI];   // 20480 B: W1 then W2, K-pair interleaved
  __shared__ float sB1[64];
  __shared__ float sB2[64];

  const int tid = threadIdx.x;
  const int rbase = blockIdx.x * 128;

  load_w_interleaved(sW, W1, tid);
  if (tid < 64) { sB1[tid] = B1[tid]; sB2[tid] = B2[tid]; }
  for (int k = tid; k < 128 * 16; k += 256) {
    int row = k >> 4, c4 = (k & 15) << 2;
    float4 v = make_float4(0.f, 0.f, 0.f, 0.f);
    int gr = rbase + row;
    if (gr < n) v = *(const float4*)(X + (size_t)gr * 64 + c4);
    *(float4*)(&sX[row * XP + c4]) = v;
  }
  __syncthreads();

  const int lane = tid & 31;
  const int w    = tid >> 5;
  const int ln   = lane & 15;
  const int hi   = lane >> 4;
  const int tr   = w * 16 + ln;          // A-fragment row owned by this lane

  // ---- GEMM1: hidden = relu(X@W1 + b1), written over this wave's own sX strip ----
  {
    v8f acc[4] = {{}, {}, {}, {}};
#pragma unroll
    for (int t = 0; t < 16; ++t) {
      v2f av = *(const v2f*)(&sX[tr * XP + 4 * t + 2 * hi]);
      const int p = (2 * t + hi) * WPI;  // K-pair row for kk = 4t+2hi
#pragma unroll
      for (int nb = 0; nb < 4; ++nb) {
        v2f bv = *(const v2f*)(&sW[p + (nb * 16 + ln) * 2]);
        acc[nb] = wmma4(av, bv, acc[nb]);
      }
    }
#pragma unroll
    for (int nb = 0; nb < 4; ++nb) {
      const int col = nb * 16 + ln;
      const float bias = sB1[col];
#pragma unroll
      for (int r = 0; r < 8; ++r) {
        float h = acc[nb][r] + bias;
        sX[(w * 16 + r + 8 * hi) * XP + col] = (h > 0.f) ? h : 0.f;
      }
    }
  }
  __syncthreads();                       // all waves done with W1
  load_w_interleaved(sW, W2, tid);
  __syncthreads();

  // ---- GEMM2: Y = hidden@W2 + b2 ----
  {
    v8f acc[4] = {{}, {}, {}, {}};
#pragma unroll
    for (int t = 0; t < 16; ++t) {
      v2f av = *(const v2f*)(&sX[tr * XP + 4 * t + 2 * hi]);
      const int p = (2 * t + hi) * WPI;
#pragma unroll
      for (int nb = 0; nb < 4; ++nb) {
        v2f bv = *(const v2f*)(&sW[p + (nb * 16 + ln) * 2]);
        acc[nb] = wmma4(av, bv, acc[nb]);
      }
    }
#pragma unroll
    for (int nb = 0; nb < 4; ++nb) {
      const int col = nb * 16 + ln;
      const float bias = sB2[col];
#pragma unroll
      for (int r = 0; r < 8; ++r) {
        int row = rbase + w * 16 + r + 8 * hi;
        if (row < n) Y[(size_t)row * 64 + col] = acc[nb][r] + bias;
      }
    }
  }
}

// -------- single linear: Y = X@W + b --------
__global__ __launch_bounds__(256) void lin_kernel(
    const float* __restrict__ X, int n,
    const float* __restrict__ W, const float* __restrict__ B,
    float* __restrict__ Y)
{
  __shared__ float sX[128 * XP];
  __shared__ float sW[32 * WPI];
  __shared__ float sB[64];

  const int tid = threadIdx.x;
  const int rbase = blockIdx.x * 128;

  load_w_interleaved(sW, W, tid);
  if (tid < 64) sB[tid] = B[tid];
  for (int k = tid; k < 128 * 16; k += 256) {
    int row = k >> 4, c4 = (k & 15) << 2;
    float4 v = make_float4(0.f, 0.f, 0.f, 0.f);
    int gr = rbase + row;
    if (gr < n) v = *(const float4*)(X + (size_t)gr * 64 + c4);
    *(float4*)(&sX[row * XP + c4]) = v;
  }
  __syncthreads();

  const int lane = tid & 31;
  const int w    = tid >> 5;
  const int ln   = lane & 15;
  const int hi   = lane >> 4;
  const int tr   = w * 16 + ln;

  v8f acc[4] = {{}, {}, {}, {}};
#pragma unroll
  for (int t = 0; t < 16; ++t) {
    v2f av = *(const v2f*)(&sX[tr * XP + 4 * t + 2 * hi]);
    const int p = (2 * t + hi) * WPI;
#pragma unroll
    for (int nb = 0; nb < 4; ++nb) {
      v2f bv = *(const v2f*)(&sW[p + (nb * 16 + ln) * 2]);
      acc[nb] = wmma4(av, bv, acc[nb]);
    }
  }
#pragma unroll
  for (int nb = 0; nb < 4; ++nb) {
    const int col = nb * 16 + ln;
    const float bias = sB[col];
#pragma unroll
    for (int r = 0; r < 8; ++r) {
      int row = rbase + w * 16 + r + 8 * hi;
      if (row < n) Y[(size_t)row * 64 + col] = acc[nb][r] + bias;
    }
  }
}

// -------- triangle gather * gather -> atomic scatter-add --------
__global__ __launch_bounds__(256) void scatter_kernel(
    const float* __restrict__ Pa, const float* __restrict__ Pb,
    const int* __restrict__ ia, const int* __restrict__ ib,
    const int* __restrict__ idst, float* __restrict__ acc, int T)
{
  long g = (long)blockIdx.x * 256 + threadIdx.x;
  int tri = (int)(g >> 4);
  if (tri >= T) return;
  int c4 = ((int)g & 15) << 2;
  float4 va = *(const float4*)(Pa + (size_t)ia[tri] * 64 + c4);
  float4 vb = *(const float4*)(Pb + (size_t)ib[tri] * 64 + c4);
  float* dst = acc + (size_t)idst[tri] * 64 + c4;
  atomicAdd(dst + 0, va.x * vb.x);
  atomicAdd(dst + 1, va.y * vb.y);
  atomicAdd(dst + 2, va.z * vb.z);
  atomicAdd(dst + 3, va.w * vb.w);
}

// -------- cur += accA + accB + accB[inv[row]], optional fused relu --------
__global__ __launch_bounds__(256) void combine_kernel(
    float* __restrict__ cur, const float* __restrict__ aA,
    const float* __restrict__ aB, const int* __restrict__ inv,
    int n, int do_relu)
{
  long g = (long)blockIdx.x * 256 + threadIdx.x;
  int row = (int)(g >> 4);
  if (row >= n) return;
  int c4 = ((int)g & 15) << 2;
  size_t o = (size_t)row * 64 + c4;
  int ir = inv[row];
  float4 v  = *(float4*)(cur + o);
  float4 va = *(const float4*)(aA + o);
  float4 vb = *(const float4*)(aB + o);
  float4 vi = *(const float4*)(aB + (size_t)ir * 64 + c4);
  v.x += va.x + vb.x + vi.x;
  v.y += va.y + vb.y + vi.y;
  v.z += va.z + vb.z + vi.z;
  v.w += va.w + vb.w + vi.w;
  if (do_relu) {
    v.x = fmaxf(v.x, 0.f); v.y = fmaxf(v.y, 0.f);
    v.z = fmaxf(v.z, 0.f); v.w = fmaxf(v.w, 0.f);
  }
  *(float4*)(cur + o) = v;
}

__global__ __launch_bounds__(256) void relu_kernel(float* __restrict__ x, long n4)
{
  long g = (long)blockIdx.x * 256 + threadIdx.x;
  if (g >= n4) return;
  float4 v = ((float4*)x)[g];
  v.x = fmaxf(v.x, 0.f); v.y = fmaxf(v.y, 0.f);
  v.z = fmaxf(v.z, 0.f); v.w = fmaxf(v.w, 0.f);
  ((float4*)x)[g] = v;
}

extern "C" void kernel_launch(void* const* d_in, const int* in_sizes, int n_in,
                              void* d_out, int out_size, void* d_ws, size_t ws_size,
                              hipStream_t stream)
{
  const float* e1   = (const float*)d_in[0];
  const float* e2   = (const float*)d_in[1];
  const int*   t111 = (const int*)d_in[2];
  const int*   t112 = (const int*)d_in[3];
  const int*   t122 = (const int*)d_in[4];
  const int*   t222 = (const int*)d_in[5];
  const int*   inv1 = (const int*)d_in[6];
  const int*   inv2 = (const int*)d_in[7];
  const float* mw1  = (const float*)d_in[8];
  const float* mb1  = (const float*)d_in[9];
  const float* mw2  = (const float*)d_in[10];
  const float* mb2  = (const float*)d_in[11];
  const float* lw1  = (const float*)d_in[12];
  const float* lb1  = (const float*)d_in[13];
  const float* lw2  = (const float*)d_in[14];
  const float* lb2  = (const float*)d_in[15];

  const int n1 = in_sizes[0] / 64;
  const int n2 = in_sizes[1] / 64;
  const int T  = in_sizes[2] / 3;
  const int L  = in_sizes[8] / (8 * 64 * 64);
  const size_t sz1 = (size_t)n1 * 64, sz2 = (size_t)n2 * 64;
  const size_t nmax = (sz1 > sz2) ? sz1 : sz2;

  // workspace layout: cur1 | cur2 | PA | PB
  float* cur1 = (float*)d_ws;
  float* cur2 = cur1 + sz1;
  float* PA   = cur2 + sz2;
  float* PB   = PA + nmax;
  // atomic accumulators live in d_out (n1==n2 in this workload); final lins overwrite it
  float* accA = (float*)d_out;
  float* accB = accA + sz1;

  hipMemcpyAsync(cur1, e1, sz1 * 4, hipMemcpyDeviceToDevice, stream);
  hipMemcpyAsync(cur2, e2, sz2 * 4, hipMemcpyDeviceToDevice, stream);

  dim3 blk(256);
  dim3 gM1((n1 + 127) / 128), gM2((n2 + 127) / 128);
  dim3 gS((unsigned)(((long)T * 16 + 255) / 256));
  dim3 gC1((unsigned)((sz1 / 4 + 255) / 256)), gC2((unsigned)((sz2 / 4 + 255) / 256));

  for (int li = 0; li < L; ++li) {
    auto w1 = [&](int mi) { return mw1 + (size_t)(li * 8 + mi) * 4096; };
    auto b1 = [&](int mi) { return mb1 + (size_t)(li * 8 + mi) * 64; };
    auto w2 = [&](int mi) { return mw2 + (size_t)(li * 8 + mi) * 4096; };
    auto b2 = [&](int mi) { return mb2 + (size_t)(li * 8 + mi) * 64; };

    // ---------- e1 update ----------
    hipMemsetAsync(accA, 0, sz1 * 4, stream);
    hipMemsetAsync(accB, 0, sz1 * 4, stream);
    // m111: mlp0(e1)[ik]*mlp0(e1)[kj] -> ij   (accA)
    mlp2_kernel<<<gM1, blk, 0, stream>>>(cur1, n1, w1(0), b1(0), w2(0), b2(0), PA);
    scatter_kernel<<<gS, blk, 0, stream>>>(PA, PA, t111 + T, t111 + 2 * (size_t)T, t111, accA, T);
    // m112: mlp1(e1)[ik]*mlp2(e2)[kj] -> ij   (accB, separate for inverse add)
    mlp2_kernel<<<gM1, blk, 0, stream>>>(cur1, n1, w1(1), b1(1), w2(1), b2(1), PA);
    mlp2_kernel<<<gM2, blk, 0, stream>>>(cur2, n2, w1(2), b1(2), w2(2), b2(2), PB);
    scatter_kernel<<<gS, blk, 0, stream>>>(PA, PB, t112 + T, t112 + 2 * (size_t)T, t112, accB, T);
    // m122: mlp3(e2)[ik]*mlp3(e2)[kj] -> ij   (accA)
    mlp2_kernel<<<gM2, blk, 0, stream>>>(cur2, n2, w1(3), b1(3), w2(3), b2(3), PA);
    scatter_kernel<<<gS, blk, 0, stream>>>(PA, PA, t122 + T, t122 + 2 * (size_t)T, t122, accA, T);
    // e1' = e1 + accA + accB + accB[inv1]   (no relu yet: mlp4/5 consume pre-relu e1')
    combine_kernel<<<gC1, blk, 0, stream>>>(cur1, accA, accB, inv1, n1, 0);

    // ---------- e2 update ----------
    hipMemsetAsync(accA, 0, sz2 * 4, stream);
    hipMemsetAsync(accB, 0, sz2 * 4, stream);
    // m211: mlp4(e1')[ij]*mlp4(e1')[ik] -> kj (accA)
    mlp2_kernel<<<gM1, blk, 0, stream>>>(cur1, n1, w1(4), b1(4), w2(4), b2(4), PA);
    scatter_kernel<<<gS, blk, 0, stream>>>(PA, PA, t112, t112 + T, t112 + 2 * (size_t)T, accA, T);
    // m212: mlp5(e1')[ij]*mlp6(e2)[kj] -> ik  (accB)
    mlp2_kernel<<<gM1, blk, 0, stream>>>(cur1, n1, w1(5), b1(5), w2(5), b2(5), PA);
    mlp2_kernel<<<gM2, blk, 0, stream>>>(cur2, n2, w1(6), b1(6), w2(6), b2(6), PB);
    scatter_kernel<<<gS, blk, 0, stream>>>(PA, PB, t122, t122 + 2 * (size_t)T, t122 + T, accB, T);
    // m222: mlp7(e2)[ik]*mlp7(e2)[kj] -> ij   (accA)
    mlp2_kernel<<<gM2, blk, 0, stream>>>(cur2, n2, w1(7), b1(7), w2(7), b2(7), PA);
    scatter_kernel<<<gS, blk, 0, stream>>>(PA, PA, t222 + T, t222 + 2 * (size_t)T, t222, accA, T);
    // e2' = relu(e2 + accA + accB + accB[inv2]);  then relu(e1')
    combine_kernel<<<gC2, blk, 0, stream>>>(cur2, accA, accB, inv2, n2, 1);
    relu_kernel<<<gC1, blk, 0, stream>>>(cur1, (long)(sz1 / 4));
  }

  // final linears overwrite d_out (acc buffers no longer needed)
  lin_kernel<<<gM1, blk, 0, stream>>>(cur1, n1, lw1, lb1, (float*)d_out);
  lin_kernel<<<gM2, blk, 0, stream>>>(cur2, n2, lw2, lb2, (float*)d_out + sz1);
}